// GIN_60997125538191
// MI455X (gfx1250) — compile-verified
//
#include <hip/hip_runtime.h>
#include <hip/hip_bf16.h>

typedef float v2f __attribute__((ext_vector_type(2)));
typedef float v8f __attribute__((ext_vector_type(8)));

#define N_NODES  100000
#define N_EDGES  1600000
#define N_GRAPHS 512
#define D_IN     128
#define D_H      64
#define D_OUT    32
#define N_CONV   4
#define BN_EPS   1e-5f

// ---------------------------------------------------------------- utilities
__global__ void zero_f32(float* __restrict__ p, int n) {
  int i = blockIdx.x * blockDim.x + threadIdx.x;
  if (i < n) p[i] = 0.f;
}

__global__ void copy_f32v4(float4* __restrict__ d, const float4* __restrict__ s, int n4) {
  int i = blockIdx.x * blockDim.x + threadIdx.x;
  if (i < n4) d[i] = s[i];
}

// -------------------------------------------------- edge scatter: agg[dst] += h[src]
// one thread per (edge, 4-feature group); d = 4 << logG
__global__ void edge_scatter(float* __restrict__ agg, const float* __restrict__ h,
                             const int* __restrict__ src, const int* __restrict__ dst,
                             int logG) {
  long long idx = (long long)blockIdx.x * blockDim.x + threadIdx.x;
  long long total = (long long)N_EDGES << logG;
  if (idx >= total) return;
  int e  = (int)(idx >> logG);
  int f4 = ((int)idx & ((1 << logG) - 1)) << 2;
  int d  = 4 << logG;
  int s  = src[e], t = dst[e];
  float4 v = *reinterpret_cast<const float4*>(h + (long long)s * d + f4);
  float* o = agg + (long long)t * d + f4;
  atomicAdd(o + 0, v.x);
  atomicAdd(o + 1, v.y);
  atomicAdd(o + 2, v.z);
  atomicAdd(o + 3, v.w);
}

// -------------------------------------------------- FP32 WMMA GEMM: C = A(MxK)@B(KxN) [+C][+bias]
// grid = (M/16, N/16), block = 32 (one wave per 16x16 tile).
// M % 16 == 0, K % 4 == 0, N % 16 == 0 guaranteed by caller.
__global__ void wmma_gemm_f32(float* __restrict__ C, const float* __restrict__ A,
                              const float* __restrict__ B, int K, int N,
                              const float* __restrict__ bias, int accumulate) {
  const int lane = threadIdx.x & 31;
  const int half = lane >> 4;    // 0: lanes 0-15, 1: lanes 16-31
  const int l16  = lane & 15;
  const int m0 = blockIdx.x << 4;
  const int n0 = blockIdx.y << 4;
  const int rowA = m0 + l16;     // A-matrix row held by this lane
  const int colB = n0 + l16;     // B/C/D column held by this lane

  v8f c = {};
  if (accumulate) {
#pragma unroll
    for (int v = 0; v < 8; ++v)
      c[v] = C[(long long)(m0 + v + 8 * half) * N + colB];
  }

  for (int k = 0; k < K; k += 4) {
    const int ka = k + 2 * half;               // this lane's K pair
    v2f a, b;
    a.x = A[(long long)rowA * K + ka];
    a.y = A[(long long)rowA * K + ka + 1];
    b.x = B[(long long)ka * N + colB];
    b.y = B[(long long)(ka + 1) * N + colB];
    // D(16x16,f32) = A(16x4,f32) x B(4x16,f32) + C
    c = __builtin_amdgcn_wmma_f32_16x16x4_f32(
        /*neg_a=*/false, a, /*neg_b=*/false, b,
        /*c_mod=*/(short)0, c, /*reuse_a=*/false, /*reuse_b=*/false);
  }

  float bv = bias ? bias[colB] : 0.f;
#pragma unroll
  for (int v = 0; v < 8; ++v)
    C[(long long)(m0 + v + 8 * half) * N + colB] = c[v] + bv;
}

// -------------------------------------------------- per-column sum / sumsq (64 cols)
// stats[0..63] = sum, stats[64..127] = sumsq. Launch total threads % 64 == 0.
__global__ void colstats64(const float* __restrict__ x, int nrows, float* __restrict__ stats) {
  int t = blockIdx.x * blockDim.x + threadIdx.x;
  int col = t & 63;
  int r = t >> 6;
  int rstep = (gridDim.x * blockDim.x) >> 6;
  float s = 0.f, q = 0.f;
  for (; r < nrows; r += rstep) {
    float v = x[((long long)r << 6) + col];
    s += v;
    q += v * v;
  }
  atomicAdd(&stats[col], s);
  atomicAdd(&stats[64 + col], q);
}

// -------------------------------------------------- BN (train-mode, biased var) + ReLU
__global__ void bn_relu64(float* __restrict__ y, const float* __restrict__ x,
                          const float* __restrict__ stats,
                          const float* __restrict__ g, const float* __restrict__ b,
                          int nrows) {
  long long i = (long long)blockIdx.x * blockDim.x + threadIdx.x;
  if (i >= (long long)nrows * 64) return;
  int col = (int)i & 63;
  float invn = 1.f / (float)nrows;
  float m = stats[col] * invn;
  float var = stats[64 + col] * invn - m * m;
  float z = (x[i] - m) * rsqrtf(var + BN_EPS) * g[col] + b[col];
  y[i] = fmaxf(z, 0.f);
}

// -------------------------------------------------- per-graph sum pooling
__global__ void pool_sum(float* __restrict__ pooled, const float* __restrict__ h,
                         const int* __restrict__ gid) {
  long long i = (long long)blockIdx.x * blockDim.x + threadIdx.x;
  if (i >= (long long)N_NODES * D_H) return;
  int node = (int)(i >> 6);
  int f = (int)i & 63;
  atomicAdd(&pooled[(long long)gid[node] * D_H + f], h[i]);
}

// ================================================================ launch
extern "C" void kernel_launch(void* const* d_in, const int* in_sizes, int n_in,
                              void* d_out, int out_size, void* d_ws, size_t ws_size,
                              hipStream_t stream) {
  (void)in_sizes; (void)n_in; (void)out_size; (void)ws_size;

  const float* h_in  = (const float*)d_in[0];
  const int*   src   = (const int*)d_in[1];
  const int*   dst   = (const int*)d_in[2];
  const int*   gid   = (const int*)d_in[3];
  const float* fc1w0 = (const float*)d_in[4];
  const float* fc1w  = (const float*)d_in[5];   // (3,64,64)
  const float* fc2w  = (const float*)d_in[6];   // (4,64,64)
  const float* bn1g  = (const float*)d_in[7];
  const float* bn1b  = (const float*)d_in[8];
  const float* bn2g  = (const float*)d_in[9];
  const float* bn2b  = (const float*)d_in[10];
  const float* predw = (const float*)d_in[11];  // (4,64,32)
  const float* predb = (const float*)d_in[12];  // (4,32)
  float* out = (float*)d_out;                   // (512,32)

  // workspace carve-out (256B aligned slices)
  char* ws = (char*)d_ws;
  size_t off = 0;
  auto take = [&](size_t bytes) -> float* {
    float* p = (float*)(ws + off);
    off += (bytes + 255) & ~(size_t)255;
    return p;
  };
  float* agg    = take((size_t)N_NODES * D_IN * 4);   // 51.2 MB (max width)
  float* t1     = take((size_t)N_NODES * D_H * 4);    // 25.6 MB
  float* t2     = take((size_t)N_NODES * D_H * 4);    // 25.6 MB
  float* hbuf   = take((size_t)N_NODES * D_H * 4);    // 25.6 MB
  float* pooled = take((size_t)N_GRAPHS * D_H * 4);   // 128 KB
  float* stats  = take((size_t)2 * D_H * 4);          // 512 B

  zero_f32<<<(N_GRAPHS * D_OUT + 255) / 256, 256, 0, stream>>>(out, N_GRAPHS * D_OUT);

  for (int l = 0; l < N_CONV; ++l) {
    const int Din = l ? D_H : D_IN;
    const float* hp = l ? hbuf : h_in;

    // agg = h + segment_sum(h[src], dst)
    int n4 = N_NODES * Din / 4;
    copy_f32v4<<<(n4 + 255) / 256, 256, 0, stream>>>((float4*)agg, (const float4*)hp, n4);
    int logG = l ? 4 : 5;                                     // Din/4 feature groups
    long long tot = (long long)N_EDGES << logG;
    edge_scatter<<<(unsigned)((tot + 255) / 256), 256, 0, stream>>>(agg, hp, src, dst, logG);

    // fc1: t1 = agg @ W1
    const float* W1 = l ? (fc1w + (size_t)(l - 1) * D_H * D_H) : fc1w0;
    dim3 gBig(N_NODES / 16, D_H / 16);
    wmma_gemm_f32<<<gBig, 32, 0, stream>>>(t1, agg, W1, Din, D_H, nullptr, 0);

    // BN1 + ReLU -> t2
    zero_f32<<<1, 128, 0, stream>>>(stats, 128);
    colstats64<<<1024, 256, 0, stream>>>(t1, N_NODES, stats);
    bn_relu64<<<(N_NODES * D_H + 255) / 256, 256, 0, stream>>>(
        t2, t1, stats, bn1g + l * D_H, bn1b + l * D_H, N_NODES);

    // fc2: t1 = t2 @ W2
    wmma_gemm_f32<<<gBig, 32, 0, stream>>>(t1, t2, fc2w + (size_t)l * D_H * D_H, D_H, D_H,
                                           nullptr, 0);

    // BN2 + ReLU -> hbuf (next layer input)
    zero_f32<<<1, 128, 0, stream>>>(stats, 128);
    colstats64<<<1024, 256, 0, stream>>>(t1, N_NODES, stats);
    bn_relu64<<<(N_NODES * D_H + 255) / 256, 256, 0, stream>>>(
        hbuf, t1, stats, bn2g + l * D_H, bn2b + l * D_H, N_NODES);

    // SumPooling per graph
    zero_f32<<<(N_GRAPHS * D_H + 255) / 256, 256, 0, stream>>>(pooled, N_GRAPHS * D_H);
    pool_sum<<<((long long)N_NODES * D_H + 255) / 256, 256, 0, stream>>>(pooled, hbuf, gid);

    // score += pooled @ pred_w[l] + pred_b[l]  (WMMA accumulates via C operand)
    dim3 gScore(N_GRAPHS / 16, D_OUT / 16);
    wmma_gemm_f32<<<gScore, 32, 0, stream>>>(out, pooled, predw + (size_t)l * D_H * D_OUT,
                                             D_H, D_OUT, predb + l * D_OUT, 1);
  }
}